// KMeansLabeller_8796093022274
// MI455X (gfx1250) — compile-verified
//
#include <hip/hip_runtime.h>

// KMeans labelling: B=256, S=128, D=32, K=8192
// labels[n] = argmin_k ||x_n - c_k||^2 = argmax_k (x_n . c_k - 0.5*||c_k||^2)
// fp32 matrix pipe: V_WMMA_F32_16X16X4_F32, two 16-point A tiles per wave
// sharing every B fragment (32x16 outputs per wave per cluster tile).

typedef __attribute__((ext_vector_type(2))) float v2f;
typedef __attribute__((ext_vector_type(8))) float v8f;

#define THREADS          128   // 4 waves
#define POINTS_PER_WAVE  32    // two 16-point tiles
#define POINTS_PER_BLOCK 128   // 4 waves * 32
#define KCHUNK           128   // clusters staged in LDS per iteration (16 KB)

__global__ __launch_bounds__(THREADS) void kmeans_label_kernel(
    const float* __restrict__ x,     // [N,32] points
    const float* __restrict__ cc,    // [K,32] centers
    float* __restrict__ out,         // [N] labels (as float)
    int K)
{
  // Swizzled B-fragment layout: float index = tile*512 + c*64 + lane*2 + sub
  // where tile = k_local/16, col = k_local%16, c = d/4, half = (d%4)/2, sub = d%2
  __shared__ float lds[KCHUNK * 32];   // 16 KB
  __shared__ float ldsb[KCHUNK];       // -0.5*||c||^2 biases

  const int tid  = threadIdx.x;
  const int wave = tid >> 5;
  const int lane = tid & 31;
  const int hlf  = lane >> 4;   // which K-pair half this lane supplies
  const int col  = lane & 15;   // M row (A) / N column (B,C,D) index

  const int pbase = blockIdx.x * POINTS_PER_BLOCK + wave * POINTS_PER_WAVE;

  // ---- A fragments for two point tiles (ISA 32-bit 16x4 A layout) ----
  v2f a0[8], a1[8];
  {
    const float* xp0 = x + (size_t)(pbase + col) * 32 + 2 * hlf;
    const float* xp1 = x + (size_t)(pbase + 16 + col) * 32 + 2 * hlf;
#pragma unroll
    for (int c = 0; c < 8; ++c) {
      a0[c] = *(const v2f*)(xp0 + 4 * c);
      a1[c] = *(const v2f*)(xp1 + 4 * c);
    }
  }

  // running argmax of g = x.c - 0.5||c||^2 ; 8 slots = 8 C/D VGPR rows
  float bv0[8], bv1[8];
  int   bi0[8], bi1[8];
#pragma unroll
  for (int v = 0; v < 8; ++v) {
    bv0[v] = -3.4e38f; bi0[v] = 0;
    bv1[v] = -3.4e38f; bi1[v] = 0;
  }

  for (int k0 = 0; k0 < K; k0 += KCHUNK) {
    __syncthreads();   // protect previous chunk's readers

    // ---- stage 128 clusters: coalesced b128 loads, swizzled LDS stores ----
    {
      const float* cptr = cc + (size_t)k0 * 32;
#pragma unroll
      for (int j = 0; j < 8; ++j) {
        int gf = tid * 4 + j * 512;         // float offset in chunk (mult of 4)
        float4 vv = *(const float4*)(cptr + gf);
        int k_local = gf >> 5;              // gf / 32
        int c       = (gf & 31) >> 2;       // (gf % 32) / 4
        float* base = &lds[(k_local >> 4) * 512 + c * 64 + (k_local & 15) * 2];
        base[0]  = vv.x;  base[1]  = vv.y;  // half 0 pair
        base[32] = vv.z;  base[33] = vv.w;  // half 1 pair
      }
    }
    __syncthreads();

    // ---- per-chunk biases from staged data (all 128 threads) ----
    {
      const float* bse = &lds[(tid >> 4) * 512 + (tid & 15) * 2];
      float s = 0.f;
#pragma unroll
      for (int c = 0; c < 8; ++c) {
        float q0 = bse[c * 64 + 0],  q1 = bse[c * 64 + 1];
        float q2 = bse[c * 64 + 32], q3 = bse[c * 64 + 33];
        s += q0 * q0 + q1 * q1 + q2 * q2 + q3 * q3;
      }
      ldsb[tid] = -0.5f * s;
    }
    __syncthreads();

    // ---- 8 cluster tiles: B fragment shared by both A tiles ----
#pragma unroll
    for (int t = 0; t < 8; ++t) {
      float bias = ldsb[t * 16 + col];
      v8f acc0 = {bias, bias, bias, bias, bias, bias, bias, bias};
      v8f acc1 = {bias, bias, bias, bias, bias, bias, bias, bias};
      const float* bbase = &lds[t * 512 + lane * 2];
#pragma unroll
      for (int c = 0; c < 8; ++c) {
        v2f bf = *(const v2f*)(bbase + c * 64);  // conflict-free ds_load_b64
        acc0 = __builtin_amdgcn_wmma_f32_16x16x4_f32(
            false, a0[c], false, bf, (short)0, acc0, false, false);
        acc1 = __builtin_amdgcn_wmma_f32_16x16x4_f32(
            false, a1[c], false, bf, (short)0, acc1, false, false);
      }
      int kidx = k0 + t * 16 + col;
#pragma unroll
      for (int v = 0; v < 8; ++v) {
        if (acc0[v] > bv0[v]) { bv0[v] = acc0[v]; bi0[v] = kidx; }
        if (acc1[v] > bv1[v]) { bv1[v] = acc1[v]; bi1[v] = kidx; }
      }
    }
  }

  // ---- cross-lane argmax reduce within 16-lane groups (ties -> lowest k) ----
#pragma unroll
  for (int v = 0; v < 8; ++v) {
    float val = bv0[v];
    int   idx = bi0[v];
#pragma unroll
    for (int m = 8; m >= 1; m >>= 1) {
      float oval = __shfl_xor(val, m, 32);
      int   oidx = __shfl_xor(idx, m, 32);
      if (oval > val || (oval == val && oidx < idx)) { val = oval; idx = oidx; }
    }
    if (col == 0) out[pbase + v + 8 * hlf] = (float)idx;
  }
#pragma unroll
  for (int v = 0; v < 8; ++v) {
    float val = bv1[v];
    int   idx = bi1[v];
#pragma unroll
    for (int m = 8; m >= 1; m >>= 1) {
      float oval = __shfl_xor(val, m, 32);
      int   oidx = __shfl_xor(idx, m, 32);
      if (oval > val || (oval == val && oidx < idx)) { val = oval; idx = oidx; }
    }
    if (col == 0) out[pbase + 16 + v + 8 * hlf] = (float)idx;
  }
}

extern "C" void kernel_launch(void* const* d_in, const int* in_sizes, int n_in,
                              void* d_out, int out_size, void* d_ws, size_t ws_size,
                              hipStream_t stream) {
  (void)n_in; (void)d_ws; (void)ws_size; (void)out_size;
  const float* x  = (const float*)d_in[0];   // [B,S,D] fp32
  const float* cc = (const float*)d_in[1];   // [K,D] fp32
  float* out = (float*)d_out;

  int npoints = in_sizes[0] / 32;            // 32768
  int K       = in_sizes[1] / 32;            // 8192
  int blocks  = npoints / POINTS_PER_BLOCK;  // 256

  kmeans_label_kernel<<<blocks, THREADS, 0, stream>>>(x, cc, out, K);
}